// optical_flow_warping_35003983462493
// MI455X (gfx1250) — compile-verified
//
#include <hip/hip_runtime.h>
#include <cstdint>

// Optical-flow bilinear warping, MI455X (gfx1250, wave32).
// Memory-bound gather: ~430 MB HBM traffic -> ~18us floor @ 23.3 TB/s.
//  - 16 threads/pixel, float4 lane-slice => each gather is one contiguous
//    256B row per half-wave (coalesced global_load_b128).
//  - Each wave owns 2 pixels and stages its own flow/mask scalars via the
//    CDNA5 async global->LDS DMA path (global_load_async_to_lds_b32,
//    ASYNCcnt) -- per-wave s_wait_asynccnt, NO block barrier needed.
//  - NT store for write-once output; L2 (192 MB) kept for the reused feature.
//  - All integer addressing via shifts (W=512 pow2, batch in blockIdx.y).

typedef float v4f __attribute__((ext_vector_type(4)));

namespace {
constexpr int B = 4, H = 384, W = 512, C = 64;
constexpr int HW = H * W;              // 196608
constexpr int PIX_PER_BLOCK = 16;
constexpr int BLOCK = 256;             // 8 waves, 2 pixels per wave
constexpr int BLOCKS_PER_IMG = HW / PIX_PER_BLOCK;  // 12288, exact
}

__global__ __launch_bounds__(BLOCK)
void warp_bilinear_kernel(const float* __restrict__ flow,   // (B,H,W,2)
                          const float* __restrict__ feat,   // (B,H,W,C)
                          const float* __restrict__ mask,   // (H,W)
                          float* __restrict__ out)          // (B,H,W,C)
{
    __shared__ float s_flow[2 * PIX_PER_BLOCK];  // 128 B: 4 floats per wave
    __shared__ float s_mask[PIX_PER_BLOCK];      //  64 B: 2 floats per wave

    const int t  = threadIdx.x;
    const int b  = blockIdx.y;
    const int p0 = blockIdx.x * PIX_PER_BLOCK;   // first pixel of this block

    const int wv = t >> 5;                 // wave id 0..7
    const int l  = t & 31;                 // lane within wave
    const int pw = p0 + (wv << 1);         // first of this wave's 2 pixels

    // ---- per-wave async global->LDS staging (6 lanes, one instruction) ----
    // lanes 0..3: this wave's 4 flow words; lanes 4..5: its 2 mask words.
    if (l < 6) {
        const float* g;
        unsigned lds;
        if (l < 4) {
            g   = flow + (((size_t)b * HW + pw) * 2 + l);
            lds = (unsigned)(uintptr_t)&s_flow[(wv << 2) + l];
        } else {
            g   = mask + (pw + (l - 4));
            lds = (unsigned)(uintptr_t)&s_mask[(wv << 1) + (l - 4)];
        }
        asm volatile("global_load_async_to_lds_b32 %0, %1, off"
                     :: "v"(lds), "v"(g) : "memory");
    }
    // ASYNCcnt is per-wave; each wave waits only for its own DMA.
    asm volatile("s_wait_asynccnt 0x0" ::: "memory");

    const int lpl = l >> 4;                // which of the wave's 2 pixels
    const int c   = (l & 15) << 2;         // channel offset: 0,4,...,60
    const int p   = pw + lpl;              // pixel within image (row-major)
    const int yg  = p >> 9;                // W = 512 (pow2)
    const int xg  = p & (W - 1);

    const float fx = s_flow[(wv << 2) + (lpl << 1) + 0];
    const float fy = s_flow[(wv << 2) + (lpl << 1) + 1];
    const float m  = s_mask[(wv << 1) + lpl];

    // Faithful to reference sequence; constant-reciprocal replaces IEEE div.
    constexpr float invWm1 = 1.0f / (float)(W - 1);
    constexpr float invHm1 = 1.0f / (float)(H - 1);
    const float xt = (float)xg + fx;
    const float yt = (float)yg + fy;
    const float x = (2.0f * xt * invWm1 - 1.0f + 1.0f) * 0.5f * (float)W;
    const float y = (2.0f * yt * invHm1 - 1.0f + 1.0f) * 0.5f * (float)H;

    int x0 = (int)floorf(x);
    x0 = min(max(x0, 0), W - 1);
    const int x1 = min(x0 + 1, W - 1);
    int y0 = (int)floorf(y);
    y0 = min(max(y0, 0), H - 1);
    const int y1 = min(y0 + 1, H - 1);

    const float x0f = (float)x0, x1f = (float)x1;
    const float y0f = (float)y0, y1f = (float)y1;
    const float wa = (x1f - x) * (y1f - y);   // pairs with (y0, x0)
    const float wb = (x1f - x) * (y - y0f);   // pairs with (y1, x0)
    const float wc = (x - x0f) * (y1f - y);   // pairs with (y0, x1)
    const float wd = (x - x0f) * (y - y0f);   // pairs with (y1, x1)

    const float* base = feat + (size_t)b * HW * C;
    const int r00 = ((y0 << 9) + x0) * C + c;   // < 12.6M, fits in int
    const int r10 = ((y1 << 9) + x0) * C + c;
    const int r01 = ((y0 << 9) + x1) * C + c;
    const int r11 = ((y1 << 9) + x1) * C + c;

    // Four independent coalesced b128 gathers in flight.
    const v4f ga = *reinterpret_cast<const v4f*>(base + r00);
    const v4f gb = *reinterpret_cast<const v4f*>(base + r10);
    const v4f gc = *reinterpret_cast<const v4f*>(base + r01);
    const v4f gd = *reinterpret_cast<const v4f*>(base + r11);

    const v4f o = (ga * wa + gb * wb + gc * wc + gd * wd) * m;

    // Write-once output -> non-temporal b128 store.
    __builtin_nontemporal_store(
        o, reinterpret_cast<v4f*>(out + ((size_t)b * HW + p) * C + c));
}

extern "C" void kernel_launch(void* const* d_in, const int* in_sizes, int n_in,
                              void* d_out, int out_size, void* d_ws, size_t ws_size,
                              hipStream_t stream) {
    const float* flow = (const float*)d_in[0];  // w:       (B,H,W,2) f32
    const float* feat = (const float*)d_in[1];  // feature: (B,H,W,C) f32
    const float* mask = (const float*)d_in[2];  // mask:    (H,W)     f32
    float* out = (float*)d_out;                 // (B,H,W,C) f32

    dim3 grid(BLOCKS_PER_IMG, B);               // (12288, 4), exact cover
    hipLaunchKernelGGL(warp_bilinear_kernel, grid, dim3(BLOCK), 0, stream,
                       flow, feat, mask, out);
}